// GRU_10007273800414
// MI455X (gfx1250) — compile-verified
//
#include <hip/hip_runtime.h>
#include <hip/hip_bf16.h>

// Problem dims
static constexpr int T_STEPS = 512;
static constexpr int BATCH   = 64;
static constexpr int DIN     = 1024;
static constexpr int HID     = 1024;

typedef float v2f __attribute__((ext_vector_type(2)));
typedef float v8f __attribute__((ext_vector_type(8)));

#define WMMA_F32(a, b, c) \
    __builtin_amdgcn_wmma_f32_16x16x4_f32(false, (a), false, (b), (short)0, (c), false, false)

// ---------------------------------------------------------------------------
// Kernel 1: zx/rx/cx = emb @ W_g_ih^T + b_g   for g in {z,r,c}
// Block: 256 threads (8 waves). Block tile: 16 rows (of T*B) x 256 cols.
// Wave tile: 16 x 32 (two 16x16 WMMA accumulators) x 3 gates.
// A-chunk (16x32) staged in LDS once per k-chunk, shared by all gates/waves.
// ---------------------------------------------------------------------------
__global__ void __launch_bounds__(256) gru_pregemm(
    const float* __restrict__ emb,
    const float* __restrict__ Wz, const float* __restrict__ Wr, const float* __restrict__ Wc,
    const float* __restrict__ bz, const float* __restrict__ br, const float* __restrict__ bc,
    float* __restrict__ zx, float* __restrict__ rx, float* __restrict__ cx)
{
    __shared__ float lA[16][32];

    const int m0   = blockIdx.x * 16;            // row tile over T*B
    const int wave = threadIdx.x >> 5;
    const int lane = threadIdx.x & 31;
    const int n0   = blockIdx.y * 256 + wave * 32;

    const int lr = lane & 15;
    const int kk = (lane < 16) ? 0 : 2;          // K sub-offset per ISA A/B layout

    v8f acc[3][2] = {};                          // [gate][n-subtile]
    const float* W[3] = {Wz, Wr, Wc};

    for (int k0 = 0; k0 < DIN; k0 += 32) {
        __syncthreads();
        {   // 256 threads x 2 floats = 16x32 chunk
            int idx = threadIdx.x * 2;
            int r = idx >> 5, c = idx & 31;
            *(v2f*)&lA[r][c] = *(const v2f*)&emb[(size_t)(m0 + r) * DIN + k0 + c];
        }
        __syncthreads();

        #pragma unroll
        for (int k4 = 0; k4 < 32; k4 += 4) {
            v2f a = *(const v2f*)&lA[lr][k4 + kk];   // A-frag shared by all 6 accums
            #pragma unroll
            for (int g = 0; g < 3; ++g) {
                #pragma unroll
                for (int s = 0; s < 2; ++s) {
                    int n = n0 + s * 16 + lr;
                    v2f b = *(const v2f*)&W[g][(size_t)n * DIN + k0 + k4 + kk];
                    acc[g][s] = WMMA_F32(a, b, acc[g][s]);
                }
            }
        }
    }

    // Epilogue: fold bias, scatter per C/D layout (VGPR i -> M = i + (lane>=16 ? 8 : 0))
    float* dst[3]        = {zx, rx, cx};
    const float* bias[3] = {bz, br, bc};
    const int mofs = (lane >= 16) ? 8 : 0;
    #pragma unroll
    for (int g = 0; g < 3; ++g) {
        #pragma unroll
        for (int s = 0; s < 2; ++s) {
            int n = n0 + s * 16 + lr;
            float bv = bias[g][n];
            #pragma unroll
            for (int i = 0; i < 8; ++i) {
                int m = m0 + i + mofs;
                dst[g][(size_t)m * HID + n] = acc[g][s][i] + bv;
            }
        }
    }
}

// ---------------------------------------------------------------------------
// Device-wide sense barrier (agent scope). Counters pre-zeroed by memset.
// ---------------------------------------------------------------------------
__device__ __forceinline__ void grid_barrier(unsigned* cnt, unsigned* gen, unsigned nblocks)
{
    __syncthreads();
    if (threadIdx.x == 0) {
        __threadfence();
        unsigned g = __hip_atomic_load(gen, __ATOMIC_ACQUIRE, __HIP_MEMORY_SCOPE_AGENT);
        unsigned prev = __hip_atomic_fetch_add(cnt, 1u, __ATOMIC_ACQ_REL, __HIP_MEMORY_SCOPE_AGENT);
        if (prev == nblocks - 1) {
            __hip_atomic_store(cnt, 0u, __ATOMIC_RELAXED, __HIP_MEMORY_SCOPE_AGENT);
            __hip_atomic_fetch_add(gen, 1u, __ATOMIC_ACQ_REL, __HIP_MEMORY_SCOPE_AGENT);
        } else {
            while (__hip_atomic_load(gen, __ATOMIC_ACQUIRE, __HIP_MEMORY_SCOPE_AGENT) == g) {
                __builtin_amdgcn_s_sleep(2);
            }
        }
    }
    __syncthreads();
}

__device__ __forceinline__ float sigmoidf_(float x) { return 1.0f / (1.0f + __expf(-x)); }

// ---------------------------------------------------------------------------
// Kernel 2: persistent recurrent scan. 64 blocks x 128 threads (4 waves).
// Block owns 16 output columns (n0), wave owns 16 batch rows.
// h_t is stored directly in out[t]; h_{t-1} read from out[t-1] (hx at t=0).
// Phase 1: z,r gates (2 WMMAs/k-step); z & h kept in VGPRs across barrier,
//          r*h spilled to ws (global A-matrix of phase 2).
// Phase 2: c gate GEMM + tanh + h update.
// ---------------------------------------------------------------------------
__global__ void __launch_bounds__(128) gru_scan(
    const float* __restrict__ hx,
    const float* __restrict__ Wz, const float* __restrict__ Wr, const float* __restrict__ Wc,
    const float* __restrict__ zx, const float* __restrict__ rx,
    float* __restrict__ out,      // [T*B*H] holds cx on entry; [+2*B*H] finals
    float* __restrict__ rh,       // [B*H] scratch
    unsigned* __restrict__ bar)   // bar[0]=cnt, bar[1]=gen (pre-zeroed)
{
    const int wave = threadIdx.x >> 5;
    const int lane = threadIdx.x & 31;
    const int n0   = blockIdx.x * 16;

    const int lr    = lane & 15;
    const int kk    = (lane < 16) ? 0 : 2;
    const int rowA  = wave * 16 + lr;                 // A-frag batch row
    const int n     = n0 + lr;                        // output column
    const int mbase = wave * 16 + ((lane >= 16) ? 8 : 0);
    const unsigned NB = gridDim.x;
    const size_t BH = (size_t)BATCH * HID;

    for (int t = 0; t < T_STEPS; ++t) {
        const float* hprev = (t == 0) ? hx : (out + (size_t)(t - 1) * BH);

        // ---- phase 1: z, r ----
        v8f az = {}, ar = {};
        for (int k0 = 0; k0 < HID; k0 += 4) {
            v2f a  = *(const v2f*)&hprev[(size_t)rowA * HID + k0 + kk];
            v2f bz = *(const v2f*)&Wz[(size_t)n * HID + k0 + kk];
            v2f br = *(const v2f*)&Wr[(size_t)n * HID + k0 + kk];
            az = WMMA_F32(a, bz, az);
            ar = WMMA_F32(a, br, ar);
        }
        float zreg[8], hvreg[8];
        #pragma unroll
        for (int i = 0; i < 8; ++i) {
            int b = mbase + i;
            size_t off = (size_t)t * BH + (size_t)b * HID + n;
            float zv = sigmoidf_(az[i] + zx[off]);    // bias folded into zx
            float rv = sigmoidf_(ar[i] + rx[off]);
            float hv = hprev[(size_t)b * HID + n];
            zreg[i] = zv;
            hvreg[i] = hv;
            rh[(size_t)b * HID + n] = rv * hv;
        }
        grid_barrier(bar, bar + 1, NB);

        // ---- phase 2: c, h update ----
        v8f ac = {};
        for (int k0 = 0; k0 < HID; k0 += 4) {
            v2f a  = *(const v2f*)&rh[(size_t)rowA * HID + k0 + kk];
            v2f bc = *(const v2f*)&Wc[(size_t)n * HID + k0 + kk];
            ac = WMMA_F32(a, bc, ac);
        }
        #pragma unroll
        for (int i = 0; i < 8; ++i) {
            int b = mbase + i;
            size_t off = (size_t)t * BH + (size_t)b * HID + n;
            float c = tanhf(ac[i] + out[off]);        // cx (+bias) pre-stored in out[t]
            out[off] = (1.0f - zreg[i]) * hvreg[i] + zreg[i] * c;
        }
        grid_barrier(bar, bar + 1, NB);
    }

    // finals: (h_T, h_T) appended after outputs
    #pragma unroll
    for (int i = 0; i < 8; ++i) {
        int b = mbase + i;
        float hv = out[(size_t)(T_STEPS - 1) * BH + (size_t)b * HID + n];
        out[(size_t)T_STEPS * BH + (size_t)b * HID + n] = hv;
        out[(size_t)T_STEPS * BH + BH + (size_t)b * HID + n] = hv;
    }
}

// ---------------------------------------------------------------------------
extern "C" void kernel_launch(void* const* d_in, const int* in_sizes, int n_in,
                              void* d_out, int out_size, void* d_ws, size_t ws_size,
                              hipStream_t stream)
{
    const float* emb  = (const float*)d_in[0];
    const float* hx   = (const float*)d_in[1];
    const float* Wzih = (const float*)d_in[2];
    const float* Wrih = (const float*)d_in[3];
    const float* Wcih = (const float*)d_in[4];
    const float* Wzhh = (const float*)d_in[5];
    const float* Wrhh = (const float*)d_in[6];
    const float* Wchh = (const float*)d_in[7];
    const float* bz   = (const float*)d_in[8];
    const float* br   = (const float*)d_in[9];
    const float* bc   = (const float*)d_in[10];
    float* out = (float*)d_out;

    const size_t TBH = (size_t)T_STEPS * BATCH * HID;
    const size_t BH  = (size_t)BATCH * HID;

    float* zx = (float*)d_ws;
    float* rx = zx + TBH;
    float* rh = rx + TBH;
    unsigned* bar = (unsigned*)(rh + BH);

    hipMemsetAsync(bar, 0, 2 * sizeof(unsigned), stream);

    // cx (+bias) goes straight into d_out[0..T*B*H)
    dim3 g1((T_STEPS * BATCH) / 16, HID / 256);
    gru_pregemm<<<g1, 256, 0, stream>>>(emb, Wzih, Wrih, Wcih, bz, br, bc, zx, rx, out);

    gru_scan<<<HID / 16, 128, 0, stream>>>(hx, Wzhh, Wrhh, Wchh, zx, rx, out, rh, bar);
}